// RTPanoNet_5669356833936
// MI455X (gfx1250) — compile-verified
//
#include <hip/hip_runtime.h>
#include <hip/hip_bf16.h>
#include <stdint.h>
#include <stddef.h>

#define NBOX 8192
#define NW   256            // 32-bit mask words per row (8192/32)
#define NMS_THRESH 0.5f

// ---------------------------------------------------------------------------
// Kernel 1: one-block bitonic sort of 64-bit keys ((~score)<<32 | idx).
// Ascending sort of key == descending score, stable by index (matches
// jnp.argsort(-scores)). Then gather sorted SoA: x1,y1,x2,y2,area,score.
// ---------------------------------------------------------------------------
__global__ __launch_bounds__(1024)
void nms_sort_kernel(const float* __restrict__ boxes,
                     const float* __restrict__ scores,
                     float* __restrict__ soa)   // [6][NBOX]
{
    extern __shared__ unsigned long long s_keys[];   // 8192 * 8B = 64 KB
    const int tid = threadIdx.x;
    const int NT  = 1024;

    for (int i = tid; i < NBOX; i += NT) {
        unsigned sb = __float_as_uint(scores[i]);    // scores in [0,1): monotone bits
        unsigned hk = 0xFFFFFFFFu - sb;              // ascending hk == descending score
        s_keys[i] = ((unsigned long long)hk << 32) | (unsigned)i;
    }
    __syncthreads();

    for (int k = 2; k <= NBOX; k <<= 1) {
        for (int j = k >> 1; j > 0; j >>= 1) {
            for (int i = tid; i < NBOX; i += NT) {
                int l = i ^ j;
                if (l > i) {
                    bool asc = ((i & k) == 0);
                    unsigned long long a = s_keys[i], b = s_keys[l];
                    if ((a > b) == asc) { s_keys[i] = b; s_keys[l] = a; }
                }
            }
            __syncthreads();
        }
    }

    for (int i = tid; i < NBOX; i += NT) {
        unsigned idx = (unsigned)(s_keys[i] & 0xFFFFFFFFu);
        float x1 = boxes[idx * 4 + 0];
        float y1 = boxes[idx * 4 + 1];
        float x2 = boxes[idx * 4 + 2];
        float y2 = boxes[idx * 4 + 3];
        soa[0 * NBOX + i] = x1;
        soa[1 * NBOX + i] = y1;
        soa[2 * NBOX + i] = x2;
        soa[3 * NBOX + i] = y2;
        soa[4 * NBOX + i] = (x2 - x1 + 1.0f) * (y2 - y1 + 1.0f);
        soa[5 * NBOX + i] = scores[idx];
    }
}

// ---------------------------------------------------------------------------
// Kernel 2: suppression bitmask. Grid (4 col-tiles, 256 row-blocks) x 256 thr.
// Each block: async-stage a 2048-column SoA tile into LDS (CDNA5
// global_load_async_to_lds_b128 + s_wait_asynccnt), then each thread owns one
// 32-column word for 8 rows; register-blocked 8x8 IoU inner kernel.
// mask[r][w] bit b set  <=>  j = 32w+b > r  and  IoU(r,j) > 0.5
// ---------------------------------------------------------------------------
__global__ __launch_bounds__(256)
void nms_mask_kernel(const float* __restrict__ soa,
                     unsigned* __restrict__ mask)
{
    extern __shared__ float s_cols[];          // [5][2048] = 40 KB
    const int tid     = threadIdx.x;
    const int tile    = blockIdx.x;            // 0..3
    const int rb      = blockIdx.y * 32;       // row base
    const int colBase = tile * 2048;

    // ---- async global -> LDS stage of 5 column arrays for this tile ----
    unsigned ldsBase = (unsigned)(uintptr_t)&s_cols[0];
    for (int a = 0; a < 5; ++a) {
        const float* gsrc = soa + a * NBOX + colBase;
        #pragma unroll
        for (int c = 0; c < 2; ++c) {
            int fi = 4 * (tid + 256 * c);                       // float idx in tile
            unsigned loff = ldsBase + (unsigned)((a * 2048 + fi) * 4);
            unsigned long long ga = (unsigned long long)(uintptr_t)(gsrc + fi);
            asm volatile("global_load_async_to_lds_b128 %0, %1, off"
                         :: "v"(loff), "v"(ga) : "memory");
        }
    }
    asm volatile("s_wait_asynccnt 0" ::: "memory");
    __syncthreads();

    const float* lx1 = s_cols;
    const float* ly1 = s_cols + 2048;
    const float* lx2 = s_cols + 2 * 2048;
    const float* ly2 = s_cols + 3 * 2048;
    const float* lar = s_cols + 4 * 2048;

    const int wl  = tid & 63;                  // local word 0..63
    const int w   = tile * 64 + wl;            // global word 0..255
    const int cl0 = wl * 32;                   // local column base

    // 8 row boxes in registers
    float bx1[8], by1[8], bx2[8], by2[8], ba[8];
    int   rrow[8];
    #pragma unroll
    for (int rr = 0; rr < 8; ++rr) {
        int r = rb + (tid >> 6) + rr * 4;
        rrow[rr] = r;
        bx1[rr] = soa[0 * NBOX + r];
        by1[rr] = soa[1 * NBOX + r];
        bx2[rr] = soa[2 * NBOX + r];
        by2[rr] = soa[3 * NBOX + r];
        ba[rr]  = soa[4 * NBOX + r];
    }

    unsigned bits[8] = {0, 0, 0, 0, 0, 0, 0, 0};
    #pragma unroll 1
    for (int bb = 0; bb < 4; ++bb) {
        // 8 column boxes in registers
        float cx1[8], cy1[8], cx2[8], cy2[8], car[8];
        #pragma unroll
        for (int u = 0; u < 8; ++u) {
            int cl = cl0 + bb * 8 + u;
            cx1[u] = lx1[cl]; cy1[u] = ly1[cl];
            cx2[u] = lx2[cl]; cy2[u] = ly2[cl];
            car[u] = lar[cl];
        }
        #pragma unroll
        for (int rr = 0; rr < 8; ++rr) {
            #pragma unroll
            for (int u = 0; u < 8; ++u) {
                int j = colBase + cl0 + bb * 8 + u;
                float ix1 = fmaxf(bx1[rr], cx1[u]);
                float iy1 = fmaxf(by1[rr], cy1[u]);
                float ix2 = fminf(bx2[rr], cx2[u]);
                float iy2 = fminf(by2[rr], cy2[u]);
                float iw  = fmaxf(ix2 - ix1 + 1.0f, 0.0f);
                float ih  = fmaxf(iy2 - iy1 + 1.0f, 0.0f);
                float inter = iw * ih;
                float iou   = inter / (ba[rr] + car[u] - inter);
                if ((iou > NMS_THRESH) && (j > rrow[rr]))
                    bits[rr] |= (1u << (bb * 8 + u));
            }
        }
    }
    #pragma unroll
    for (int rr = 0; rr < 8; ++rr)
        mask[(size_t)rrow[rr] * NW + w] = bits[rr];
}

// ---------------------------------------------------------------------------
// Kernel 3: sequential greedy scan (single block, 256 threads) + finalize.
// Per 32-row chunk c: wave0 resolves intra-chunk suppression from diagonal
// words via v_readlane, then every thread ORs surviving rows' mask words into
// its register-resident `removed` word. Prefetch next chunk into L2.
// ---------------------------------------------------------------------------
__global__ __launch_bounds__(256)
void nms_reduce_kernel(const unsigned* __restrict__ mask,
                       const float* __restrict__ soa,
                       float* __restrict__ out)
{
    __shared__ unsigned s_remword;
    __shared__ unsigned s_alive;
    __shared__ unsigned s_keep[NW];
    const int t = threadIdx.x;
    unsigned removed = 0u;

    #pragma unroll 1
    for (int c = 0; c < NW; ++c) {
        if (t == c) s_remword = removed;      // removed word for this chunk's cols
        __syncthreads();

        if (t < 32) {                         // wave0: intra-chunk resolution
            unsigned sup   = mask[(size_t)(32 * c + t) * NW + c]; // diagonal word
            unsigned alive = ~s_remword;
            #pragma unroll 1
            for (int b = 0; b < 32; ++b) {
                if ((alive >> b) & 1u)
                    alive &= ~(unsigned)__builtin_amdgcn_readlane((int)sup, b);
            }
            if (t == 0) s_alive = alive;
        }
        if (c + 1 < NW)                       // warm L2 for next chunk's rows
            __builtin_prefetch(&mask[(size_t)(32 * (c + 1) + (t & 31)) * NW + t], 0, 1);
        __syncthreads();

        unsigned alive = s_alive;
        #pragma unroll 1
        for (int b = 0; b < 32; ++b) {
            if ((alive >> b) & 1u)
                removed |= mask[(size_t)(32 * c + b) * NW + t];
        }
        __syncthreads();
    }

    s_keep[t] = ~removed;
    __syncthreads();

    float* boxes_out  = out;                  // 8192 x 4
    float* scores_out = out + NBOX * 4;       // 8192
    float* keep_out   = out + NBOX * 5;       // 8192 (bool as 0/1 float)
    for (int i = t; i < NBOX; i += 256) {
        float m = (float)((s_keep[i >> 5] >> (i & 31)) & 1u);
        boxes_out[i * 4 + 0] = soa[0 * NBOX + i] * m;
        boxes_out[i * 4 + 1] = soa[1 * NBOX + i] * m;
        boxes_out[i * 4 + 2] = soa[2 * NBOX + i] * m;
        boxes_out[i * 4 + 3] = soa[3 * NBOX + i] * m;
        scores_out[i] = soa[5 * NBOX + i] * m;
        keep_out[i]   = m;
    }
}

// ---------------------------------------------------------------------------
extern "C" void kernel_launch(void* const* d_in, const int* in_sizes, int n_in,
                              void* d_out, int out_size, void* d_ws, size_t ws_size,
                              hipStream_t stream) {
    const float* boxes  = (const float*)d_in[0];   // (8192, 4)
    const float* scores = (const float*)d_in[1];   // (8192,)
    float*    out  = (float*)d_out;                // 49152 floats
    float*    soa  = (float*)d_ws;                 // 6*8192 floats (192 KB)
    unsigned* mask = (unsigned*)((char*)d_ws + (256u << 10)); // 8 MB @ +256 KB

    nms_sort_kernel<<<1, 1024, NBOX * sizeof(unsigned long long), stream>>>(
        boxes, scores, soa);

    dim3 g2(4, 256);   // 4 column tiles x 256 row blocks
    nms_mask_kernel<<<g2, 256, 5 * 2048 * sizeof(float), stream>>>(soa, mask);

    nms_reduce_kernel<<<1, 256, 0, stream>>>(mask, soa, out);
}